// Sinkhorn_77154792505448
// MI455X (gfx1250) — compile-verified
//
#include <hip/hip_runtime.h>
#include <stdint.h>
#include <math.h>

// Sinkhorn [64,1024,1024] f32, TAU=0.01, 5 iters, out = exp(y)+1e-6.
// Dual-potential formulation: y = x/TAU + u[b,i] + v[b,j].
//   row half-step: u[b,i] = -LSE_j(x/TAU + v[b,j])
//   col half-step: v[b,j] = -LSE_i(x/TAU + u[b,i])
// Each pass reads x once (256MB) and writes 256KB. Serpentine batch order
// between passes exploits the 192MB L2 (working set 256MB).

#define NROWS   65536              // 64 * 1024
#define NCOLS   1024
#define INV_TAU (1.0f / 0.01f)
#define EPSV    1e-6f

// ---------- CDNA5 async global->LDS (ASYNCcnt) helpers ----------
typedef int v4i __attribute__((vector_size(4 * sizeof(int))));
typedef __attribute__((address_space(1))) v4i as1_v4i;   // global (prints as __device__)
typedef __attribute__((address_space(3))) v4i as3_v4i;   // LDS    (prints as __shared__)

__device__ __forceinline__ void cp_async_b128(const float* g, float* lds) {
#if __has_builtin(__builtin_amdgcn_global_load_async_to_lds_b128)
  __builtin_amdgcn_global_load_async_to_lds_b128(
      (as1_v4i*)(uintptr_t)g,
      (as3_v4i*)(uint32_t)(uintptr_t)lds,
      0, 0);
#else
  uint32_t loff = (uint32_t)(uintptr_t)lds;
  uint64_t gaddr = (uint64_t)(uintptr_t)g;
  asm volatile("global_load_async_to_lds_b128 %0, %1, off"
               :: "v"(loff), "v"(gaddr) : "memory");
#endif
}

__device__ __forceinline__ void wait_async_le1() {
#if __has_builtin(__builtin_amdgcn_s_wait_asynccnt)
  __builtin_amdgcn_s_wait_asynccnt(1);
  asm volatile("" ::: "memory");   // compiler ordering barrier
#else
  asm volatile("s_wait_asynccnt 0x1" ::: "memory");
#endif
}

__device__ __forceinline__ void wait_async_le0() {
#if __has_builtin(__builtin_amdgcn_s_wait_asynccnt)
  __builtin_amdgcn_s_wait_asynccnt(0);
  asm volatile("" ::: "memory");
#else
  asm volatile("s_wait_asynccnt 0x0" ::: "memory");
#endif
}

__device__ __forceinline__ void wait_ds0() {
#if __has_builtin(__builtin_amdgcn_s_wait_dscnt)
  __builtin_amdgcn_s_wait_dscnt(0);
  asm volatile("" ::: "memory");
#else
  asm volatile("s_wait_dscnt 0x0" ::: "memory");
#endif
}

// ---------- online log-sum-exp primitives (single exp per element) ----------
__device__ __forceinline__ void lse_acc(float& m, float& s, float t) {
  if (t > m) { s = s * __expf(m - t) + 1.0f; m = t; }   // exp(-inf)=0 on first hit
  else       { s += __expf(t - m); }
}
__device__ __forceinline__ void lse_merge(float& m, float& s, float m2, float s2) {
  float M = fmaxf(m, m2);
  s = s * __expf(m - M) + s2 * __expf(m2 - M);
  m = M;
}

// ---------- kernels ----------
__global__ __launch_bounds__(256) void zero_kernel(float* __restrict__ p, int n) {
  int i = blockIdx.x * 256 + threadIdx.x;
  if (i < n) p[i] = 0.0f;
}

// One wave per row; async double-buffered 512B chunks through LDS.
// grid = 8192 blocks x 256 threads (8 waves -> 8 rows per block).
__global__ __launch_bounds__(256) void row_pass(const float* __restrict__ x,
                                                const float* __restrict__ v,
                                                float* __restrict__ u, int rev) {
  __shared__ float stage[2][8][128];                 // 2 bufs x 8 waves x 512B
  int eb  = rev ? ((int)gridDim.x - 1 - (int)blockIdx.x) : (int)blockIdx.x;
  int w   = threadIdx.x >> 5;
  int l   = threadIdx.x & 31;
  int row = eb * 8 + w;
  int b   = row >> 10;
  const float* rowp = x + ((size_t)row << 10);
  const float* vp   = v + ((size_t)b << 10);
  float* l0 = &stage[0][w][l * 4];
  float* l1 = &stage[1][w][l * 4];
  int c0 = l * 4;

  cp_async_b128(rowp + c0,       l0);                // chunk 0
  cp_async_b128(rowp + c0 + 128, l1);                // chunk 1

  float m = -INFINITY, s = 0.0f;
#pragma unroll
  for (int c = 0; c < 8; ++c) {
    if (c < 7) wait_async_le1(); else wait_async_le0();
    float* lp = (c & 1) ? l1 : l0;
    float4 vv = *reinterpret_cast<const float4*>(vp + c * 128 + c0);
    float a0 = lp[0], a1 = lp[1], a2 = lp[2], a3 = lp[3];
    lse_acc(m, s, a0 * INV_TAU + vv.x);
    lse_acc(m, s, a1 * INV_TAU + vv.y);
    lse_acc(m, s, a2 * INV_TAU + vv.z);
    lse_acc(m, s, a3 * INV_TAU + vv.w);
    if (c < 6) {                                     // refill consumed buffer
      wait_ds0();                                    // LDS reads done before overwrite
      cp_async_b128(rowp + c0 + (c + 2) * 128, lp);
    }
  }

  // wave32 (m,s) LSE tree-merge
#pragma unroll
  for (int off = 16; off; off >>= 1) {
    float m2 = __shfl_xor(m, off, 32);
    float s2 = __shfl_xor(s, off, 32);
    lse_merge(m, s, m2, s2);
  }
  if (l == 0) u[row] = -(m + __logf(s));
}

// Column partial LSE: grid = 1024 blocks (b x 4 row-chunks x 4 col-groups),
// thread t owns one column, fully coalesced b32 streams.
__global__ __launch_bounds__(256) void col_pass_part(const float* __restrict__ x,
                                                     const float* __restrict__ u,
                                                     float2* __restrict__ part, int rev) {
  int eb = rev ? ((int)gridDim.x - 1 - (int)blockIdx.x) : (int)blockIdx.x;
  int b  = eb >> 4;
  int rc = (eb >> 2) & 3;
  int cg = eb & 3;
  int j  = cg * 256 + threadIdx.x;
  const float* xp = x + ((size_t)b << 20) + ((size_t)(rc * 256) << 10) + j;
  const float* up = u + (b << 10) + rc * 256;        // uniform -> SMEM loads
  float m = -INFINITY, s = 0.0f;
  for (int i = 0; i < 256; i += 4) {
    float u0 = up[i + 0], u1 = up[i + 1], u2 = up[i + 2], u3 = up[i + 3];
    float t0 = xp[(size_t)(i + 0) << 10] * INV_TAU + u0;
    float t1 = xp[(size_t)(i + 1) << 10] * INV_TAU + u1;
    float t2 = xp[(size_t)(i + 2) << 10] * INV_TAU + u2;
    float t3 = xp[(size_t)(i + 3) << 10] * INV_TAU + u3;
    lse_acc(m, s, t0); lse_acc(m, s, t1);
    lse_acc(m, s, t2); lse_acc(m, s, t3);
  }
  part[((size_t)(b * 4 + rc) << 10) + j] = make_float2(m, s);
}

// Merge 4 row-chunk partials per column -> v. grid = 256 x 256.
__global__ __launch_bounds__(256) void col_pass_comb(const float2* __restrict__ part,
                                                     float* __restrict__ v) {
  int idx = blockIdx.x * 256 + threadIdx.x;          // b*1024 + j
  int b = idx >> 10, j = idx & 1023;
  float m = -INFINITY, s = 0.0f;
#pragma unroll
  for (int rc = 0; rc < 4; ++rc) {
    float2 p = part[((size_t)(b * 4 + rc) << 10) + j];
    lse_merge(m, s, p.x, p.y);
  }
  v[idx] = -(m + __logf(s));
}

// out = exp(x/TAU + u_i + v_j) + EPS. grid = 65536 blocks (one row each).
__global__ __launch_bounds__(256) void final_pass(const float* __restrict__ x,
                                                  const float* __restrict__ u,
                                                  const float* __restrict__ v,
                                                  float* __restrict__ out) {
  int row = blockIdx.x;
  int b   = row >> 10;
  size_t base = (size_t)row << 10;
  int j = threadIdx.x << 2;
  float ui = u[row];                                 // uniform -> SMEM load
  float4 xv = *reinterpret_cast<const float4*>(x + base + j);
  float4 vv = *reinterpret_cast<const float4*>(v + ((size_t)b << 10) + j);
  float4 o;
  o.x = __expf(xv.x * INV_TAU + ui + vv.x) + EPSV;
  o.y = __expf(xv.y * INV_TAU + ui + vv.y) + EPSV;
  o.z = __expf(xv.z * INV_TAU + ui + vv.z) + EPSV;
  o.w = __expf(xv.w * INV_TAU + ui + vv.w) + EPSV;
  *reinterpret_cast<float4*>(out + base + j) = o;
}

// ---------- launcher ----------
extern "C" void kernel_launch(void* const* d_in, const int* in_sizes, int n_in,
                              void* d_out, int out_size, void* d_ws, size_t ws_size,
                              hipStream_t stream) {
  (void)in_sizes; (void)n_in; (void)out_size; (void)ws_size;
  const float* x = (const float*)d_in[0];
  float* out = (float*)d_out;
  // workspace: u[65536] | v[65536] | part[64*4*1024] float2  (2.5 MB total)
  float*  u    = (float*)d_ws;
  float*  v    = u + NROWS;
  float2* part = (float2*)(v + NROWS);

  zero_kernel<<<256, 256, 0, stream>>>(v, NROWS);    // v = 0 for first row pass

  for (int it = 0; it < 5; ++it) {
    // serpentine batch order: row passes forward, col passes backward -> L2 reuse
    row_pass<<<8192, 256, 0, stream>>>(x, v, u, 0);
    col_pass_part<<<1024, 256, 0, stream>>>(x, u, part, 1);
    col_pass_comb<<<256, 256, 0, stream>>>(part, v);
  }
  final_pass<<<65536, 256, 0, stream>>>(x, u, v, out);
}